// RoiPoolingConv_88862873354942
// MI455X (gfx1250) — compile-verified
//
#include <hip/hip_runtime.h>
#include <hip/hip_bf16.h>
#include <stdint.h>

// ROI bilinear pool: img (1,128,128,512) f32, rois (512,4) i32 [x,y,w,h]
// out (1,512,14,14,512) f32.
// Memory-bound: ~237 MB HBM traffic -> ~10us floor @ 23.3 TB/s.
// Strategy: one 128-thread block per (roi, py) row; double-buffered
// async-to-LDS staging of the 4 corner channel-vectors (CDNA5 ASYNCcnt path),
// blend from LDS, stream results out with non-temporal stores so the 32MB
// image stays resident in the 192MB L2.

typedef float v4f __attribute__((ext_vector_type(4)));

#define POOL  14
#define NROI  512
#define IMG_W 128
#define CH    512

__device__ __forceinline__ void async_load_16B(const float* sbase,
                                               unsigned voff_bytes,
                                               unsigned lds_bytes) {
  // GVS form: mem_addr = SADDR64 + VADDR32 ; VDST VGPR holds LDS byte address.
  asm volatile("global_load_async_to_lds_b128 %0, %1, %2"
               :: "v"(lds_bytes), "v"(voff_bytes), "s"(sbase)
               : "memory");
}

// TF2 bilinear half-pixel mapping (matches reference):
// v = (p+0.5)*size/P - 0.5 ; t = v - floor(v) (unclamped floor);
// c0 = clip(floor(v),0,size-1), c1 = clip(ceil(v),0,size-1)
__device__ __forceinline__ void xline(int p, float sizef, int r0,
                                      int* c0, int* c1, float* t) {
  float v = ((float)p + 0.5f) * (sizef * (1.0f / (float)POOL)) - 0.5f;
  float f = floorf(v);
  *t = v - f;
  float m = sizef - 1.0f;
  *c0 = (int)fminf(fmaxf(f, 0.0f), m) + r0;
  *c1 = (int)fminf(fmaxf(ceilf(v), 0.0f), m) + r0;
}

__global__ __launch_bounds__(128) void roi_pool_async(
    const float* __restrict__ img, const int* __restrict__ rois,
    float* __restrict__ out) {
  // double buffer x 4 corners x 512 channels = 16 KB
  __shared__ __align__(16) float lbuf[2 * 4 * CH];

  const int tid = threadIdx.x;           // 0..127, float4 per thread = 512 ch
  const unsigned ch = (unsigned)tid * 4u;

  const int r  = blockIdx.x / POOL;
  const int py = blockIdx.x % POOL;

  const int rx = rois[r * 4 + 0];
  const int ry = rois[r * 4 + 1];
  const float wf = (float)rois[r * 4 + 2];
  const float hf = (float)rois[r * 4 + 3];

  int y0, y1;
  float ly;
  xline(py, hf, ry, &y0, &y1, &ly);

  const unsigned rowbytes = IMG_W * CH * 4u;
  const unsigned y0b = (unsigned)y0 * rowbytes;
  const unsigned y1b = (unsigned)y1 * rowbytes;

  // True LDS byte offset of lbuf within this block's LDS allocation.
  const unsigned lbase = (unsigned)(unsigned long long)(
      (__attribute__((address_space(3))) float*)lbuf);

  auto prefetch = [&](int px, int buf) {
    int x0, x1;
    float lx;
    xline(px, wf, rx, &x0, &x1, &lx);
    unsigned x0b = (unsigned)x0 * (CH * 4u) + ch * 4u;
    unsigned x1b = (unsigned)x1 * (CH * 4u) + ch * 4u;
    unsigned lb = lbase + (unsigned)buf * (4u * CH * 4u) + ch * 4u;
    async_load_16B(img, y0b + x0b, lb + 0u * CH * 4u);
    async_load_16B(img, y0b + x1b, lb + 1u * CH * 4u);
    async_load_16B(img, y1b + x0b, lb + 2u * CH * 4u);
    async_load_16B(img, y1b + x1b, lb + 3u * CH * 4u);
  };

  prefetch(0, 0);

  for (int px = 0; px < POOL; ++px) {
    const int cur = px & 1;
    if (px + 1 < POOL) {
      prefetch(px + 1, cur ^ 1);  // 8 outstanding async loads in this wave
      // wait until only the 4 just-issued remain -> current buffer complete
      asm volatile("s_wait_asynccnt 4" ::: "memory");
    } else {
      asm volatile("s_wait_asynccnt 0" ::: "memory");
    }

    int x0, x1;
    float lx;
    xline(px, wf, rx, &x0, &x1, &lx);

    const float* lb = lbuf + cur * (4 * CH) + ch;
    v4f c00 = *(const v4f*)(lb + 0 * CH);
    v4f c01 = *(const v4f*)(lb + 1 * CH);
    v4f c10 = *(const v4f*)(lb + 2 * CH);
    v4f c11 = *(const v4f*)(lb + 3 * CH);

    v4f top = c00 * (1.0f - lx) + c01 * lx;
    v4f bot = c10 * (1.0f - lx) + c11 * lx;
    v4f res = top * (1.0f - ly) + bot * ly;

    // Streaming store: output is write-once, keep L2 for the image.
    size_t off = ((size_t)blockIdx.x * POOL + (size_t)px) * CH + ch;
    __builtin_nontemporal_store(res, (v4f*)(out + off));
  }
}

extern "C" void kernel_launch(void* const* d_in, const int* in_sizes, int n_in,
                              void* d_out, int out_size, void* d_ws,
                              size_t ws_size, hipStream_t stream) {
  const float* img = (const float*)d_in[0];   // (1,128,128,512) f32
  const int* rois  = (const int*)d_in[1];     // (512,4) i32
  float* out = (float*)d_out;                 // (1,512,14,14,512) f32

  dim3 grid(NROI * POOL);  // one block per (roi, output row)
  dim3 block(128);
  hipLaunchKernelGGL(roi_pool_async, grid, block, 0, stream, img, rois, out);
}